// TGCN_16243566313999
// MI455X (gfx1250) — compile-verified
//
#include <hip/hip_runtime.h>
#include <hip/hip_bf16.h>

typedef float v2f __attribute__((ext_vector_type(2)));
typedef float v8f __attribute__((ext_vector_type(8)));

// ---------------------------------------------------------------- zero fill
__global__ void zero_f32(float* __restrict__ p, long n) {
  long i = (long)blockIdx.x * blockDim.x + threadIdx.x;
  if (i < n) p[i] = 0.0f;
}

// ------------------------------------------------- fused-concat edge scatter
// Wave-cooperative: one wave per edge, 32 lanes split the feature dimension.
// out[dst*C + c] += w_e * feat(src, c), C = C1 + H
//   c <  C1 : feat = pA[src*lda + c*sA]           (x slice or h0)
//   c >= C1 : feat = h[src*H + (c-C1)]            (optionally * r gate)
// r gate: rz[src*2H + (c-C1)]  (first half of rz == r)
__global__ void scatter_gconv(const int* __restrict__ src, const int* __restrict__ dst,
                              const float* __restrict__ w, int E,
                              const float* __restrict__ pA, int lda, int sA, int C1,
                              const float* __restrict__ ph, int H,
                              const float* __restrict__ rz,
                              float* __restrict__ out) {
  const int lane = threadIdx.x & 31;
  const int e = blockIdx.x * (blockDim.x >> 5) + (threadIdx.x >> 5);
  if (e >= E) return;                          // wave-uniform

  // wave-uniform edge data: force into SGPRs so address math is scalar
  const int   s  = __builtin_amdgcn_readfirstlane(src[e]);
  const int   d  = __builtin_amdgcn_readfirstlane(dst[e]);
  const float we = __uint_as_float(__builtin_amdgcn_readfirstlane(__float_as_uint(w[e])));

  const int C = C1 + H;
  float* __restrict__ o = out + (long)d * C;

  // part A (x slice: short/strided, or h0: contiguous)
  const float* __restrict__ ar = pA + (long)s * lda;
  for (int c = lane; c < C1; c += 32)
    atomicAdd(&o[c], we * ar[c * sA]);

  // part B: hidden state, contiguous -> coalesced loads & dense atomic bursts
  const float* __restrict__ hr = ph + (long)s * H;
  if (rz != nullptr) {
    const float* __restrict__ rr = rz + (long)s * (2 * H);
    for (int c = lane; c < H; c += 32)
      atomicAdd(&o[C1 + c], we * (rr[c] * hr[c]));
  } else {
    for (int c = lane; c < H; c += 32)
      atomicAdd(&o[C1 + c], we * hr[c]);
  }
}

// ------------------------------------------------------ WMMA f32 tile GEMM
// out(N,M) = act(A(N,K) @ W(K,M) + bias), one wave per 16x16 tile.
// ACT: 0 = sigmoid, 1 = tanh
// A frag (16x4 f32, ISA 7.12.2): lane m = lane&15, k = k0 + 2*(lane>>4) + j
// B frag (4x16 f32):             lane n = lane&15, k = k0 + 2*(lane>>4) + j
// D frag (16x16 f32):            m = v + 8*(lane>>4), n = lane&15
template <int ACT>
__global__ void gemm16_wmma(const float* __restrict__ A, const float* __restrict__ W,
                            const float* __restrict__ bias, float* __restrict__ out,
                            int N, int K, int M) {
  const int lane   = threadIdx.x & 31;
  const int waveIb = threadIdx.x >> 5;
  const int tileRow = blockIdx.x * (blockDim.x >> 5) + waveIb;
  const int tileCol = blockIdx.y;
  if (tileRow * 16 >= N) return;          // wave-uniform: EXEC stays all-ones

  const int mi = lane & 15;
  const int kHalf = lane >> 4;            // 0 or 1
  const int rowA = min(tileRow * 16 + mi, N - 1);
  const float* __restrict__ Arow = A + (long)rowA * K;
  const float* __restrict__ Wcol = W + (tileCol * 16 + mi); // mi doubles as n for B

  v8f acc = {};
  for (int k0 = 0; k0 < K; k0 += 4) {
    v2f a, b;
    a.x = Arow[k0 + 2 * kHalf + 0];
    a.y = Arow[k0 + 2 * kHalf + 1];
    b.x = Wcol[(long)(k0 + 2 * kHalf + 0) * M];
    b.y = Wcol[(long)(k0 + 2 * kHalf + 1) * M];
    acc = __builtin_amdgcn_wmma_f32_16x16x4_f32(
        false, a, false, b, (short)0, acc, false, false);
  }

  const int n = lane & 15;
  const int colg = tileCol * 16 + n;
  const float bv = bias[colg];
#pragma unroll
  for (int v = 0; v < 8; ++v) {
    const int mm = v + 8 * (lane >> 4);
    const int rowg = tileRow * 16 + mm;
    if (rowg < N) {
      float xv = acc[v] + bv;
      if (ACT == 0)       xv = 1.0f / (1.0f + expf(-xv));   // sigmoid
      else if (ACT == 1)  xv = tanhf(xv);
      out[(long)rowg * M + colg] = xv;
    }
  }
}

// ----------------------------------------------------------- GRU state mix
// h = z*h + (1-z)*c,  z = rz[:, H:2H]
__global__ void gru_update(const float* __restrict__ rz, const float* __restrict__ c,
                           float* __restrict__ h, int N, int H) {
  long i = (long)blockIdx.x * blockDim.x + threadIdx.x;
  if (i >= (long)N * H) return;
  int node = (int)(i / H), j = (int)(i % H);
  float z = rz[(long)node * 2 * H + H + j];
  h[i] = z * h[i] + (1.0f - z) * c[i];
}

// --------------------------------------------------------- final projection
__global__ void out_proj(const float* __restrict__ h, const float* __restrict__ Wout,
                         const float* __restrict__ bout, float* __restrict__ out,
                         int N, int H, int OUT) {
  int i = blockIdx.x * blockDim.x + threadIdx.x;
  if (i >= N * OUT) return;
  int node = i / OUT, o = i % OUT;
  float acc = bout[o];
  const float* hr = h + (long)node * H;
  for (int j = 0; j < H; ++j) acc += hr[j] * Wout[j * OUT + o];
  out[i] = acc;
}

// ------------------------------------------------------------------ driver
template <int ACT>
static void launch_gemm(const float* A, int K, const float* W, const float* b,
                        float* out, int M, int N, hipStream_t stream) {
  const int wavesPerBlock = 4;
  const int rowTiles = (N + 15) / 16;
  dim3 grid((rowTiles + wavesPerBlock - 1) / wavesPerBlock, M / 16);
  gemm16_wmma<ACT><<<grid, dim3(32 * wavesPerBlock), 0, stream>>>(A, W, b, out, N, K, M);
}

extern "C" void kernel_launch(void* const* d_in, const int* in_sizes, int n_in,
                              void* d_out, int out_size, void* d_ws, size_t ws_size,
                              hipStream_t stream) {
  const float* x    = (const float*)d_in[0];
  const int*   eidx = (const int*)  d_in[1];
  const float* ew   = (const float*)d_in[2];
  const float* Wg0  = (const float*)d_in[3];
  const float* bg0  = (const float*)d_in[4];
  const float* Wc0  = (const float*)d_in[5];
  const float* bc0  = (const float*)d_in[6];
  const float* Wg1  = (const float*)d_in[7];
  const float* bg1  = (const float*)d_in[8];
  const float* Wc1  = (const float*)d_in[9];
  const float* bc1  = (const float*)d_in[10];
  const float* Wout = (const float*)d_in[11];
  const float* bout = (const float*)d_in[12];

  const int E    = in_sizes[2];
  const int twoH = in_sizes[4];           // bg0 = 2H
  const int H    = twoH / 2;
  const int FH   = in_sizes[3] / twoH;    // Wg0 = (F+H, 2H)
  const int F    = FH - H;
  const int OUT  = in_sizes[12];          // bout
  const int N    = out_size / OUT;
  const int T    = in_sizes[0] / (N * F); // x = (N, F, T)

  const int* src = eidx;
  const int* dst = eidx + E;

  float* ws  = (float*)d_ws;
  float* h0  = ws;                         // N*H
  float* h1  = h0  + (long)N * H;          // N*H
  float* agg = h1  + (long)N * H;          // N*2H (holds K<=2H aggregation)
  float* rz  = agg + (long)N * twoH;       // N*2H
  float* cb  = rz  + (long)N * twoH;       // N*H

  auto zero = [&](float* p, long n) {
    zero_f32<<<dim3((unsigned)((n + 255) / 256)), dim3(256), 0, stream>>>(p, n);
  };

  zero(h0, (long)N * H);
  zero(h1, (long)N * H);

  const int wavesPerBlk = 8;               // 256 threads, one wave per edge
  const int eBlocks = (E + wavesPerBlk - 1) / wavesPerBlk;
  const int updBlocks = (int)(((long)N * H + 255) / 256);

  for (int t = 0; t < T; ++t) {
    const float* xt = x + t;               // element (node,f) = xt[node*F*T + f*T]
    // ---------------- layer 0: x_t -> h0 ----------------
    const int C0 = F + H;
    zero(agg, (long)N * C0);
    scatter_gconv<<<eBlocks, 256, 0, stream>>>(src, dst, ew, E,
                                               xt, F * T, T, F, h0, H, nullptr, agg);
    launch_gemm<0>(agg, C0, Wg0, bg0, rz, twoH, N, stream);       // rz = sigmoid
    zero(agg, (long)N * C0);
    scatter_gconv<<<eBlocks, 256, 0, stream>>>(src, dst, ew, E,
                                               xt, F * T, T, F, h0, H, rz, agg);
    launch_gemm<1>(agg, C0, Wc0, bc0, cb, H, N, stream);          // c = tanh
    gru_update<<<updBlocks, 256, 0, stream>>>(rz, cb, h0, N, H);
    // ---------------- layer 1: h0 -> h1 ----------------
    const int C1 = H + H;
    zero(agg, (long)N * C1);
    scatter_gconv<<<eBlocks, 256, 0, stream>>>(src, dst, ew, E,
                                               h0, H, 1, H, h1, H, nullptr, agg);
    launch_gemm<0>(agg, C1, Wg1, bg1, rz, twoH, N, stream);
    zero(agg, (long)N * C1);
    scatter_gconv<<<eBlocks, 256, 0, stream>>>(src, dst, ew, E,
                                               h0, H, 1, H, h1, H, rz, agg);
    launch_gemm<1>(agg, C1, Wc1, bc1, cb, H, N, stream);
    gru_update<<<updBlocks, 256, 0, stream>>>(rz, cb, h1, N, H);
  }

  out_proj<<<(N * OUT + 255) / 256, 256, 0, stream>>>(h1, Wout, bout,
                                                      (float*)d_out, N, H, OUT);
}